// GraphSAGEBinary_33440615367324
// MI455X (gfx1250) — compile-verified
//
#include <hip/hip_runtime.h>

typedef __attribute__((ext_vector_type(2))) float v2f;
typedef __attribute__((ext_vector_type(8))) float v8f;

#define D 128            // IN_DIM == HIDDEN == 128
#define EPB 128          // edges per scatter block

// ---------------------------------------------------------------------------
// Degree count: count[dst] += 1
// ---------------------------------------------------------------------------
__global__ void k_count(const int* __restrict__ dst, float* __restrict__ cnt, int E) {
    int e = blockIdx.x * blockDim.x + threadIdx.x;
    if (e < E) atomicAdd(&cnt[dst[e]], 1.0f);
}

// ---------------------------------------------------------------------------
// Scatter-add of source features onto dst rows.
// blockDim.x == 128: thread t handles feature t (fully coalesced 512B per
// edge for both the gather and the atomic).  Edge indices staged in LDS.
// ---------------------------------------------------------------------------
__global__ void k_scatter(const float* __restrict__ x, const int* __restrict__ src,
                          const int* __restrict__ dst, float* __restrict__ aggr, int E) {
    __shared__ int s_src[EPB];
    __shared__ int s_dst[EPB];
    const int t    = threadIdx.x;
    const int base = blockIdx.x * EPB;
    if (base + t < E) { s_src[t] = src[base + t]; s_dst[t] = dst[base + t]; }
    __syncthreads();
    const int nE = min(EPB, E - base);
    for (int i = 0; i < nE; ++i) {
        float v = x[(size_t)s_src[i] * D + t];
        atomicAdd(&aggr[(size_t)s_dst[i] * D + t], v);
    }
}

// ---------------------------------------------------------------------------
// fp32 WMMA (16x16x4) tiled GEMM, compile-time specialized:
//   DUAL=1 : out = epi( (A1 @ W1.T) * inv_deg? + A2 @ W2.T + bias )
//   DUAL=0 : out = epi(  A1 @ W1.T + bias )
// SCALE=1 : first product scaled per output row by 1/max(cnt[row],1)
//           (mean aggregation, hoisted out of the K loop by linearity).
// Block: 256 threads = 8 waves; one 16-row M tile per block; wave w covers
// output cols [16w, 16w+16).  K loop processed in 16-wide chunks: all
// fragment loads for a chunk are issued before its 4 (or 8) WMMAs so
// multiple global_load_b64 stay in flight per s_wait_loadcnt.
// Fragment layouts per ISA 7.12.2:
//   A 16x4 : lane(lo) = row, VGPR0 = K{kk}, VGPR1 = K{kk+1}, kk = k + 2*hi
//   B 4x16 : lane(lo) = col, same K split
//   C/D    : VGPR r = row mbase + r + 8*hi, col nbase + lo
// ---------------------------------------------------------------------------
template <int DUAL, int SCALE, int RELU>
__global__ void k_gemm(const float* __restrict__ A1, const float* __restrict__ cnt,
                       const float* __restrict__ W1, const float* __restrict__ A2,
                       const float* __restrict__ W2, const float* __restrict__ bias,
                       float* __restrict__ out) {
    const int lane  = threadIdx.x & 31;
    const int wave  = threadIdx.x >> 5;
    const int lo    = lane & 15;
    const int hi    = lane >> 4;
    const int mbase = blockIdx.x * 16;
    const int nbase = wave * 16;
    const int m     = mbase + lo;         // A-fragment row for this lane
    const int n     = nbase + lo;         // B-fragment col / store col

    const float* __restrict__ a1row = A1 + (size_t)m * D;
    const float* __restrict__ w1row = W1 + (size_t)n * D;
    const float* __restrict__ a2row = DUAL ? A2 + (size_t)m * D : nullptr;
    const float* __restrict__ w2row = DUAL ? W2 + (size_t)n * D : nullptr;

    v8f acc1 = {};
    v8f acc2 = {};

    #pragma unroll
    for (int k0 = 0; k0 < D; k0 += 16) {
        v2f a1f[4], b1f[4], a2f[4], b2f[4];
        // ---- issue every load of this chunk before any WMMA consumes them
        #pragma unroll
        for (int j = 0; j < 4; ++j) {
            const int kk = k0 + 4 * j + 2 * hi;
            a1f[j].x = a1row[kk];     a1f[j].y = a1row[kk + 1];
            b1f[j].x = w1row[kk];     b1f[j].y = w1row[kk + 1];
            if (DUAL) {
                a2f[j].x = a2row[kk]; a2f[j].y = a2row[kk + 1];
                b2f[j].x = w2row[kk]; b2f[j].y = w2row[kk + 1];
            }
        }
        // ---- matrix ops (two independent accumulators -> no RAW chain
        //      between consecutive WMMAs of the same iteration)
        #pragma unroll
        for (int j = 0; j < 4; ++j) {
            acc1 = __builtin_amdgcn_wmma_f32_16x16x4_f32(false, a1f[j], false, b1f[j],
                                                         (short)0, acc1, false, false);
            if (DUAL)
                acc2 = __builtin_amdgcn_wmma_f32_16x16x4_f32(false, a2f[j], false, b2f[j],
                                                             (short)0, acc2, false, false);
        }
    }

    const float bn = bias[n];
    #pragma unroll
    for (int r = 0; r < 8; ++r) {
        const int mm = mbase + r + 8 * hi;      // C/D row for register r
        float v = acc1[r];
        if (SCALE) v *= 1.0f / fmaxf(cnt[mm], 1.0f);   // mean aggregation
        if (DUAL)  v += acc2[r];
        v += bn;
        if (RELU)  v = fmaxf(v, 0.0f);
        out[(size_t)mm * D + nbase + lo] = v;
    }
}

// ---------------------------------------------------------------------------
// logits[n] = h[n,:] . mw2[0,:] + mb2
// ---------------------------------------------------------------------------
__global__ void k_logits(const float* __restrict__ h, const float* __restrict__ w,
                         const float* __restrict__ b, float* __restrict__ out, int N) {
    int nidx = blockIdx.x * blockDim.x + threadIdx.x;
    if (nidx >= N) return;
    const float4* row = (const float4*)(h + (size_t)nidx * D);
    const float4* wv  = (const float4*)w;
    float acc = 0.0f;
    #pragma unroll
    for (int k = 0; k < D / 4; ++k) {
        float4 a = row[k];
        float4 c = wv[k];
        acc += a.x * c.x + a.y * c.y + a.z * c.z + a.w * c.w;
    }
    out[nidx] = acc + b[0];
}

// ---------------------------------------------------------------------------
extern "C" void kernel_launch(void* const* d_in, const int* in_sizes, int n_in,
                              void* d_out, int out_size, void* d_ws, size_t ws_size,
                              hipStream_t stream) {
    const float* x   = (const float*)d_in[0];
    const int*   ei  = (const int*)d_in[1];     // [2, E]
    const float* w1l = (const float*)d_in[2];
    const float* b1l = (const float*)d_in[3];
    const float* w1r = (const float*)d_in[4];
    const float* w2l = (const float*)d_in[5];
    const float* b2l = (const float*)d_in[6];
    const float* w2r = (const float*)d_in[7];
    const float* mw1 = (const float*)d_in[8];
    const float* mb1 = (const float*)d_in[9];
    const float* mw2 = (const float*)d_in[10];
    const float* mb2 = (const float*)d_in[11];

    const int N = in_sizes[0] / D;              // 50000
    const int E = in_sizes[1] / 2;              // 1600000
    const int* src = ei;
    const int* dst = ei + E;

    const size_t feat = (size_t)N * D;
    float* bufA = (float*)d_ws;                 // aggr (layers 1 & 2), then h3
    float* bufB = bufA + feat;                  // h1
    float* bufC = bufB + feat;                  // h2
    float* cnt  = bufC + feat;                  // degree counts [N]

    const dim3 sBlk(EPB), sGrd((E + EPB - 1) / EPB);
    const dim3 gBlk(256), gGrd(N / 16);         // 50000 / 16 = 3125

    // degree counts (same graph both layers)
    hipMemsetAsync(cnt, 0, (size_t)N * sizeof(float), stream);
    k_count<<<(E + 255) / 256, 256, 0, stream>>>(dst, cnt, E);

    // ---- layer 1: h1 = relu(mean_aggr(x) @ w1l.T + b1l + x @ w1r.T)
    hipMemsetAsync(bufA, 0, feat * sizeof(float), stream);
    k_scatter<<<sGrd, sBlk, 0, stream>>>(x, src, dst, bufA, E);
    k_gemm<1, 1, 1><<<gGrd, gBlk, 0, stream>>>(bufA, cnt, w1l, x, w1r, b1l, bufB);

    // ---- layer 2: h2 = relu(mean_aggr(h1) @ w2l.T + b2l + h1 @ w2r.T)
    hipMemsetAsync(bufA, 0, feat * sizeof(float), stream);
    k_scatter<<<sGrd, sBlk, 0, stream>>>(bufB, src, dst, bufA, E);
    k_gemm<1, 1, 1><<<gGrd, gBlk, 0, stream>>>(bufA, cnt, w2l, bufB, w2r, b2l, bufC);

    // ---- MLP: h3 = relu(h2 @ mw1.T + mb1)
    k_gemm<0, 0, 1><<<gGrd, gBlk, 0, stream>>>(bufC, nullptr, mw1, nullptr, nullptr, mb1, bufA);

    // ---- logits = h3 @ mw2.T + mb2
    k_logits<<<(N + 255) / 256, 256, 0, stream>>>(bufA, mw2, mb2, (float*)d_out, N);
}